// PeriodicAttentionHead_60361470378379
// MI455X (gfx1250) — compile-verified
//
#include <hip/hip_runtime.h>
#include <math.h>
#include <stdint.h>

#define Bn   8
#define Nn   512
#define Dn   28
#define HIDn 64
#define VSTR 68   // padded LDS row stride (floats): 272B => 4-bank advance/lane, conflict-free b128
#define RSTR 68

typedef __attribute__((ext_vector_type(2))) float v2f;
typedef __attribute__((ext_vector_type(8))) float v8f;

__device__ __forceinline__ float gelu_exact(float x) {
    // exact GELU (approximate=False): 0.5*x*(1+erf(x/sqrt(2)))
    return 0.5f * x * (1.0f + erff(x * 0.70710678118654752440f));
}

// ---------------- Precompute U,V (per (b,row)) ----------------
__global__ __launch_bounds__(64) void uv_kernel(const float* __restrict__ feat,
                                                const float* __restrict__ W1,
                                                const float* __restrict__ b1,
                                                float* __restrict__ Ug,
                                                float* __restrict__ Vg) {
    int row = blockIdx.x;          // 0 .. B*N-1
    int h   = threadIdx.x;         // 0 .. 63
    const float* f = feat + (size_t)row * Dn;
    float u = b1[h];
    float v = 0.0f;
#pragma unroll
    for (int d = 0; d < Dn; ++d) {
        float fd = f[d];
        u = fmaf(fd, W1[d * HIDn + h], u);                 // feat_i block
        v = fmaf(fd, W1[(Dn + d) * HIDn + h], v);          // feat_j block
    }
    v = fmaf(f[9], W1[62 * HIDn + h], v);                  // barrier_j term
    Ug[(size_t)row * HIDn + h] = u;
    Vg[(size_t)row * HIDn + h] = v;
}

// ---------------- Precompute R[rel] table (rel = -511..511) ----------------
__global__ __launch_bounds__(64) void rel_kernel(const float* __restrict__ W1,
                                                 const float* __restrict__ dist_table,
                                                 float* __restrict__ Rg) {
    int rr = blockIdx.x;           // 0..1022 -> rel = rr-511
    int h  = threadIdx.x;
    int rel  = rr - 511;
    int dist = rel < 0 ? -rel : rel;
    if (dist > Nn - 1) dist = Nn - 1;
    float acc = 0.0f;
#pragma unroll
    for (int k = 0; k < 4; ++k)
        acc = fmaf(dist_table[dist * 4 + k], W1[(56 + k) * HIDn + h], acc);
    acc = fmaf((float)rel * (1.0f / (float)Nn), W1[60 * HIDn + h], acc);
    acc = fmaf(1.0f / ((float)dist + 1.0f),     W1[61 * HIDn + h], acc);
    Rg[(size_t)rr * HIDn + h] = acc;
}

// ------- async global->LDS staging (CDNA5 GLOBAL_LOAD_ASYNC_TO_LDS_B128) -------
// Copies `rows` rows of 64 floats from contiguous src into LDS with row stride
// `dstr` floats (padded). One async b128 per 16B; tracked by ASYNCcnt.
__device__ __forceinline__ void cp_lds_async(float* dst_base, int dstr,
                                             const float* __restrict__ src, int rows,
                                             int tid, int nthr) {
    uint32_t base = (uint32_t)(uintptr_t)dst_base;   // low 32b of LDS aperture == LDS offset
    int total = rows * 16;                           // 16 x float4 per 64-float row
    for (int k = tid; k < total; k += nthr) {
        int row = k >> 4;
        int c   = k & 15;
        uint32_t ldsb = base + (uint32_t)(row * dstr + c * 4) * 4u;
        const float* g = src + (size_t)row * HIDn + c * 4;
        asm volatile("global_load_async_to_lds_b128 %0, %1, off"
                     :
                     : "v"(ldsb), "v"(g)
                     : "memory");
    }
}

// ---------------- Main: WMMA dot tiles + fused pairwise MLP ----------------
// grid.x = B * 32 * 2  (b, i-tile, j-split); block = 256 (8 wave32)
__global__ __launch_bounds__(256) void score_kernel(const float* __restrict__ feat,
                                                    const float* __restrict__ W1,
                                                    const float* __restrict__ W2,
                                                    const float* __restrict__ b2,
                                                    const float* __restrict__ Ug,
                                                    const float* __restrict__ Vg,
                                                    const float* __restrict__ Rg,
                                                    float* __restrict__ out) {
    __shared__ float lU[16 * HIDn];        //   4 KB (reads are lane-broadcast)
    __shared__ float lV[Nn * VSTR];        // ~136 KB, padded stride
    __shared__ float lR[527 * RSTR];       // ~140 KB, padded stride
    __shared__ float lW[2 * HIDn];         // W1 row 63 + W2

    int bidx     = blockIdx.x;
    int half_blk = bidx & 1;
    int it       = (bidx >> 1) & 31;
    int b        = bidx >> 6;
    int i0       = it * 16;
    int tid      = threadIdx.x;

    // Stage only the causally-needed prefixes of V and R via the async engine.
    cp_lds_async(lU, HIDn, Ug + (size_t)(b * Nn + i0) * HIDn, 16, tid, 256);
    cp_lds_async(lV, VSTR, Vg + (size_t)b * Nn * HIDn, i0 + 16, tid, 256);
    cp_lds_async(lR, RSTR, Rg + (size_t)(496 - i0) * HIDn, i0 + 31, tid, 256);
    if (tid < HIDn) {
        lW[tid]        = W1[63 * HIDn + tid];   // dot-term weight row
        lW[HIDn + tid] = W2[tid];
    }
    asm volatile("s_wait_asynccnt 0" ::: "memory");
    __syncthreads();

    int wave  = tid >> 5;
    int lane  = tid & 31;
    int n     = lane & 15;     // column within 16x16 tile (j) / A row (i)
    int hi    = lane >> 4;     // lane-half selects K pair for A/B fragments
    int mbase = hi * 8;        // C rows this lane holds: mbase .. mbase+7
    float b2v = b2[0];
    const float inv_sqrt_d = 0.18898223650461363f;   // 1/sqrt(28)

    // A fragments: Fi tile (16 x 28 f32), fixed for the whole block.
    v2f a[7];
    {
        const float* fi = feat + (size_t)(b * Nn + i0 + n) * Dn;
#pragma unroll
        for (int kc = 0; kc < 7; ++kc) {
            int c0 = kc * 4 + 2 * hi;
            v2f t = { fi[c0], fi[c0 + 1] };
            a[kc] = t;
        }
    }

    int cstart = wave + 8 * half_blk;            // 0..15
    for (int jt = cstart; jt <= it; jt += 16) {  // causal: only j0 <= i0+15
        int j0 = jt * 16;

        // B fragments: Fj^T tile (28 x 16) — same addressing pattern as A.
        v2f bm[7];
        const float* fj = feat + (size_t)(b * Nn + j0 + n) * Dn;
#pragma unroll
        for (int kc = 0; kc < 7; ++kc) {
            int c0 = kc * 4 + 2 * hi;
            v2f t = { fj[c0], fj[c0 + 1] };
            bm[kc] = t;
        }

        // dot tile = Fi @ Fj^T via exact-fp32 WMMA, K=28 in 7 steps of 4
        v8f acc = { 0.f, 0.f, 0.f, 0.f, 0.f, 0.f, 0.f, 0.f };
#pragma unroll
        for (int kc = 0; kc < 7; ++kc)
            acc = __builtin_amdgcn_wmma_f32_16x16x4_f32(
                false, a[kc], false, bm[kc], (short)0, acc, false, false);

        float sdot[8], sc[8];
#pragma unroll
        for (int r = 0; r < 8; ++r) { sdot[r] = acc[r] * inv_sqrt_d; sc[r] = 0.0f; }

        int j = j0 + n;                              // this lane's j (fixed)
        const float* vrow = lV + (size_t)j * VSTR;   // reused across 8 i's
        int rbase = j - mbase + 15;                  // lR row for r is rbase-r

        for (int h4 = 0; h4 < HIDn; h4 += 4) {
            float4 v4 = *(const float4*)(vrow + h4);
            float4 wa = *(const float4*)(lW + h4);
            float4 wb = *(const float4*)(lW + HIDn + h4);
#pragma unroll
            for (int r = 0; r < 8; ++r) {
                float4 u4 = *(const float4*)(lU + (mbase + r) * HIDn + h4);
                float4 r4 = *(const float4*)(lR + (size_t)(rbase - r) * RSTR + h4);
                float z0 = fmaf(sdot[r], wa.x, u4.x + v4.x + r4.x);
                float z1 = fmaf(sdot[r], wa.y, u4.y + v4.y + r4.y);
                float z2 = fmaf(sdot[r], wa.z, u4.z + v4.z + r4.z);
                float z3 = fmaf(sdot[r], wa.w, u4.w + v4.w + r4.w);
                sc[r] = fmaf(gelu_exact(z0), wb.x, sc[r]);
                sc[r] = fmaf(gelu_exact(z1), wb.y, sc[r]);
                sc[r] = fmaf(gelu_exact(z2), wb.z, sc[r]);
                sc[r] = fmaf(gelu_exact(z3), wb.w, sc[r]);
            }
        }

#pragma unroll
        for (int r = 0; r < 8; ++r) {
            int i = i0 + mbase + r;
            float s = (j <= i) ? (sc[r] + b2v) : -__builtin_inff();
            out[((size_t)b * Nn + i) * Nn + j] = s;
        }
    }
}

// ---------------- Row softmax: one wave32 per (b,i) row ----------------
__global__ __launch_bounds__(256) void softmax_kernel(float* __restrict__ out) {
    int wave = threadIdx.x >> 5;
    int lane = threadIdx.x & 31;
    int row  = blockIdx.x * 8 + wave;       // 0 .. B*N-1
    int i    = row & (Nn - 1);
    float* p = out + (size_t)row * Nn;
    float s[16];
    float mx = -__builtin_inff();
#pragma unroll
    for (int k = 0; k < 16; ++k) {
        int j = k * 32 + lane;
        float v = (j <= i) ? p[j] : -__builtin_inff();  // never read poison
        s[k] = v;
        mx = fmaxf(mx, v);
    }
#pragma unroll
    for (int off = 16; off > 0; off >>= 1) mx = fmaxf(mx, __shfl_xor(mx, off, 32));
    float sum = 0.0f;
#pragma unroll
    for (int k = 0; k < 16; ++k) { float e = expf(s[k] - mx); s[k] = e; sum += e; }
#pragma unroll
    for (int off = 16; off > 0; off >>= 1) sum += __shfl_xor(sum, off, 32);
    float inv = 1.0f / sum;
#pragma unroll
    for (int k = 0; k < 16; ++k) p[k * 32 + lane] = s[k] * inv;   // zeros for j>i
}

extern "C" void kernel_launch(void* const* d_in, const int* in_sizes, int n_in,
                              void* d_out, int out_size, void* d_ws, size_t ws_size,
                              hipStream_t stream) {
    const float* feat       = (const float*)d_in[0];
    // d_in[1] = mask (bool tril) — causal mask is computed analytically
    const float* W1         = (const float*)d_in[2];
    const float* b1         = (const float*)d_in[3];
    const float* W2         = (const float*)d_in[4];
    const float* b2         = (const float*)d_in[5];
    const float* dist_table = (const float*)d_in[6];

    float* Ug = (float*)d_ws;                          // B*N*64 floats (1 MB)
    float* Vg = Ug + (size_t)Bn * Nn * HIDn;           // B*N*64 floats (1 MB)
    float* Rg = Vg + (size_t)Bn * Nn * HIDn;           // 1023*64 floats (~256 KB)
    float* out = (float*)d_out;

    uv_kernel<<<Bn * Nn, 64, 0, stream>>>(feat, W1, b1, Ug, Vg);
    rel_kernel<<<2 * Nn - 1, 64, 0, stream>>>(W1, dist_table, Rg);
    score_kernel<<<Bn * 32 * 2, 256, 0, stream>>>(feat, W1, W2, b2, Ug, Vg, Rg, out);
    softmax_kernel<<<(Bn * Nn) / 8, 256, 0, stream>>>(out);
}